// IndexKernel_32238024524411
// MI455X (gfx1250) — compile-verified
//
#include <hip/hip_runtime.h>

// CDNA5 (gfx1250) kernel for batched low-rank covariance lookups:
//   out[p] = sum_c  dot(cf[c, x[p,c], :16], cf[c, y[p,c], :16])
//          + sum_c  (x[p,c]==y[p,c] ? std[c, x[p,c]]^2 : 0)
//
// Strategy: one wave32 handles 16 pairs. For each column c and each K-block
// of 4, issue V_WMMA_F32_16X16X4_F32 with A = cf[x-rows] (16x4) and
// B = cf[y-rows]^T (4x16); 16 chained WMMAs accumulate the full 16-long dot
// for all (x_m, y_n) combinations; the per-pair results are diag(D).
// Gather traffic (~0.5 GB of 64B rows over a 2MB table) is served by L2;
// HBM traffic is ~38 MB (x, y, out) -> ~1.6 us roofline at 23.3 TB/s.

typedef float v2f __attribute__((ext_vector_type(2)));
typedef float v8f __attribute__((ext_vector_type(8)));
typedef int   v4i __attribute__((ext_vector_type(4)));

#define NB_CAT_LOG2 13      // 8192 categories
#define N_PAIRS     1048576
#define PAIRS_PER_WAVE 16
#define WAVES_PER_BLOCK 8
#define BLOCK_THREADS (WAVES_PER_BLOCK * 32)
#define GRID_BLOCKS (N_PAIRS / (PAIRS_PER_WAVE * WAVES_PER_BLOCK))  // 8192

__global__ __launch_bounds__(BLOCK_THREADS)
void covpair_wmma_kernel(const int* __restrict__ x,
                         const int* __restrict__ y,
                         const float* __restrict__ cf,    // [4][8192][16]
                         const float* __restrict__ stdv,  // [4][8192]
                         float* __restrict__ out)         // [N_PAIRS]
{
    const unsigned lane = threadIdx.x & 31u;
    const unsigned wave = blockIdx.x * WAVES_PER_BLOCK + (threadIdx.x >> 5);
    const int      P    = (int)(wave * PAIRS_PER_WAVE);  // base pair index
    const unsigned m    = lane & 15u;                    // pair-in-wave / N index
    const unsigned h    = lane >> 4;                     // K half selector (0/1)

    // Per-lane index rows (lanes m and m+16 redundantly load the same row;
    // b128, fully coalesced, L2-friendly).
    const v4i xr = *(const v4i*)(x + (size_t)(P + (int)m) * 4);
    const v4i yr = *(const v4i*)(y + (size_t)(P + (int)m) * 4);

    v8f acc = {0.f, 0.f, 0.f, 0.f, 0.f, 0.f, 0.f, 0.f};
    float stdc = 0.f;

#pragma unroll
    for (int c = 0; c < 4; ++c) {
        const int xi = xr[c];
        const int yi = yr[c];

        // Row base + A/B fragment offset for this lane:
        //   A 16x4 f32 layout: lane m+16h holds {K=2h, K=2h+1} per K-block.
        //   B 4x16  f32 layout: lane n+16h holds {B[2h][n], B[2h+1][n]} --
        //   same float2-at-(4*kb+2h) gather, just from the y-row.
        const float* rowx = cf + ((((size_t)c << NB_CAT_LOG2) + (size_t)xi) << 4) + (h << 1);
        const float* rowy = cf + ((((size_t)c << NB_CAT_LOG2) + (size_t)yi) << 4) + (h << 1);

#pragma unroll
        for (int kb = 0; kb < 4; ++kb) {
            v2f a = *(const v2f*)(rowx + (kb << 2));
            v2f b = *(const v2f*)(rowy + (kb << 2));
            // D = A*B + C, f32 16x16x4; accumulate across kb and c.
            acc = __builtin_amdgcn_wmma_f32_16x16x4_f32(
                /*neg_a=*/false, a, /*neg_b=*/false, b,
                /*c_mod=*/(short)0, acc, /*reuse_a=*/false, /*reuse_b=*/false);
        }

        // Diagonal std^2 correction, branch-free so EXEC stays all-ones
        // around every WMMA (ISA requirement). std table is 128 KB -> L2 hits.
        const float s = stdv[((size_t)c << NB_CAT_LOG2) + (size_t)xi];
        stdc += (xi == yi) ? s * s : 0.f;
    }

    // Extract diag(D): pair m lives in lane m (m<8, elem m) or lane m+16
    // (m>=8, elem m-8). 3-level v_cndmask select of acc[lane & 7].
    float t0 = (lane & 1u) ? acc[1] : acc[0];
    float t1 = (lane & 1u) ? acc[3] : acc[2];
    float t2 = (lane & 1u) ? acc[5] : acc[4];
    float t3 = (lane & 1u) ? acc[7] : acc[6];
    float u0 = (lane & 2u) ? t1 : t0;
    float u1 = (lane & 2u) ? t3 : t2;
    float d  = (lane & 4u) ? u1 : u0;

    // Valid diagonal holders: lanes 0..7 (pairs 0..7) and 24..31 (pairs 8..15).
    if (lane < 8u || lane >= 24u) {
        out[P + (int)m] = d + stdc;
    }
}

extern "C" void kernel_launch(void* const* d_in, const int* in_sizes, int n_in,
                              void* d_out, int out_size, void* d_ws, size_t ws_size,
                              hipStream_t stream) {
    (void)in_sizes; (void)n_in; (void)d_ws; (void)ws_size; (void)out_size;
    const int*   x    = (const int*)d_in[0];
    const int*   y    = (const int*)d_in[1];
    const float* cf   = (const float*)d_in[2];
    const float* stdv = (const float*)d_in[3];
    float*       out  = (float*)d_out;

    covpair_wmma_kernel<<<GRID_BLOCKS, BLOCK_THREADS, 0, stream>>>(x, y, cf, stdv, out);
}